// GAT_10642928959815
// MI455X (gfx1250) — compile-verified
//
#include <hip/hip_runtime.h>
#include <hip/hip_bf16.h>
#include <math.h>

#define DIM_IN  64
#define HID     128
#define HEADS   8
#define DIM_H   16
#define DIM_OUT 64

typedef __attribute__((ext_vector_type(2))) float v2f;
typedef __attribute__((ext_vector_type(8))) float v8f;

// ---------- helpers ----------
__device__ __forceinline__ void atomicMaxFloat(float* addr, float val) {
  // IEEE trick: non-negative floats order as signed ints, negative as reversed unsigned.
  if (val >= 0.0f) atomicMax((int*)addr, __float_as_int(val));
  else             atomicMin((unsigned int*)addr, __float_as_uint(val));
}

// Non-temporal (TH=NT) accessors for stream-once data: keep the 192MB L2 free
// for the randomly gathered node tables, which have ~30x reuse per cache line.
__device__ __forceinline__ int   ntload(const int* p)   { return __builtin_nontemporal_load(p); }
__device__ __forceinline__ float ntload(const float* p) { return __builtin_nontemporal_load(p); }
__device__ __forceinline__ void  ntstore(float* p, float v) { __builtin_nontemporal_store(v, p); }

__global__ void fill_kernel(float* __restrict__ p, float v, int n) {
  int i = blockIdx.x * blockDim.x + threadIdx.x;
  if (i < n) p[i] = v;
}

// ---------- self-loop edge-attr mean ----------
__global__ void degree_kernel(const int* __restrict__ dst, const float* __restrict__ ew,
                              float* __restrict__ cnt, float* __restrict__ sm, int E) {
  int e = blockIdx.x * blockDim.x + threadIdx.x;
  if (e >= E) return;
  int d = ntload(dst + e);
  atomicAdd(&cnt[d], 1.0f);
  atomicAdd(&sm[d], ntload(ew + e));
}

__global__ void mean_kernel(const float* __restrict__ cnt, const float* __restrict__ sm,
                            float* __restrict__ mean, int n) {
  int i = blockIdx.x * blockDim.x + threadIdx.x;
  if (i < n) mean[i] = (cnt[i] > 0.0f) ? (sm[i] / cnt[i]) : 0.0f;
}

// ---------- fp32 WMMA GEMM: Out[M,NOUT] = A[M,K] @ W[K,NOUT] + bias ----------
// blockDim.x = 32*(NOUT/16); one wave per 16-col tile, 16 rows per block.
template<int K, int NOUT>
__global__ void gemm_wmma_kernel(const float* __restrict__ A, const float* __restrict__ W,
                                 const float* __restrict__ bias, float* __restrict__ Out, int M) {
  const int lane = threadIdx.x & 31;
  const int wv   = threadIdx.x >> 5;
  const int col  = wv * 16 + (lane & 15);
  const int row0 = blockIdx.x * 16;
  const int m    = lane & 15;
  const int hi   = lane >> 4;            // 0: K pair {0,1}, 1: K pair {2,3}
  int ra = row0 + m; if (ra > M - 1) ra = M - 1;   // clamp tail rows (EXEC stays all-1)
  const float* __restrict__ Arow = A + (size_t)ra * K;
  v8f acc = {0.f, 0.f, 0.f, 0.f, 0.f, 0.f, 0.f, 0.f};
  #pragma unroll
  for (int k0 = 0; k0 < K; k0 += 4) {
    const int kA = k0 + hi * 2;
    v2f a; a.x = Arow[kA];                      a.y = Arow[kA + 1];
    v2f b; b.x = W[(size_t)kA * NOUT + col];    b.y = W[(size_t)(kA + 1) * NOUT + col];
    acc = __builtin_amdgcn_wmma_f32_16x16x4_f32(false, a, false, b,
                                                (short)0, acc, false, false);
  }
  const float bv = bias[col];
  #pragma unroll
  for (int v = 0; v < 8; ++v) {
    const int r = row0 + v + hi * 8;
    if (r < M) Out[(size_t)r * NOUT + col] = acc[v] + bv;
  }
}

// ---------- layer 1 (H=8, C=16): wave per edge ----------
__global__ void l1_logits_kernel(const float* __restrict__ xl, const float* __restrict__ xr,
                                 const float* __restrict__ w1e, const float* __restrict__ att1,
                                 const int* __restrict__ src, const int* __restrict__ dst,
                                 const float* __restrict__ ew, const float* __restrict__ mean_attr,
                                 float* __restrict__ logits, float* __restrict__ lmax,
                                 int E, int Etot) {
  const int lane = threadIdx.x & 31;
  const int edge = blockIdx.x * (blockDim.x >> 5) + (threadIdx.x >> 5);
  if (edge >= Etot) return;
  int s, d; float attr;
  if (edge < E) { s = ntload(src + edge); d = ntload(dst + edge); attr = ntload(ew + edge); }
  else          { s = edge - E;  d = s;   attr = mean_attr[s]; }
  const float4 a  = ((const float4*)(xl + (size_t)s * HID))[lane];
  const float4 b  = ((const float4*)(xr + (size_t)d * HID))[lane];
  const float4 e4 = ((const float4*)w1e)[lane];
  const float4 at = ((const float4*)att1)[lane];
  float m0 = a.x + b.x + e4.x * attr; m0 = (m0 > 0.f) ? m0 : 0.2f * m0;
  float m1 = a.y + b.y + e4.y * attr; m1 = (m1 > 0.f) ? m1 : 0.2f * m1;
  float m2 = a.z + b.z + e4.z * attr; m2 = (m2 > 0.f) ? m2 : 0.2f * m2;
  float m3 = a.w + b.w + e4.w * attr; m3 = (m3 > 0.f) ? m3 : 0.2f * m3;
  float p = m0 * at.x + m1 * at.y + m2 * at.z + m3 * at.w;
  p += __shfl_xor(p, 1, 32);      // 4 lanes = 16 channels = one head
  p += __shfl_xor(p, 2, 32);
  if ((lane & 3) == 0) {
    const int h = lane >> 2;
    ntstore(logits + (size_t)edge * HEADS + h, p);
    atomicMaxFloat(&lmax[(size_t)d * HEADS + h], p);
  }
}

__global__ void l1_exp_kernel(float* __restrict__ logits, const float* __restrict__ lmax,
                              float* __restrict__ denom, const int* __restrict__ dst,
                              int E, int Etot) {
  const int i = blockIdx.x * blockDim.x + threadIdx.x;
  if (i >= Etot * HEADS) return;
  const int e = i >> 3, h = i & 7;
  const int d = (e < E) ? ntload(dst + e) : (e - E);
  const float ex = expf(ntload(logits + i) - lmax[(size_t)d * HEADS + h]);
  ntstore(logits + i, ex);
  atomicAdd(&denom[(size_t)d * HEADS + h], ex);
}

__global__ void l1_aggregate_kernel(const float* __restrict__ xl, const float* __restrict__ logits,
                                    const float* __restrict__ denom,
                                    const int* __restrict__ src, const int* __restrict__ dst,
                                    float* __restrict__ out, int E, int Etot) {
  const int lane = threadIdx.x & 31;
  const int edge = blockIdx.x * (blockDim.x >> 5) + (threadIdx.x >> 5);
  if (edge >= Etot) return;
  int s, d;
  if (edge < E) { s = ntload(src + edge); d = ntload(dst + edge); } else { s = edge - E; d = s; }
  const int h = lane >> 2;
  const float alpha = ntload(logits + (size_t)edge * HEADS + h) / denom[(size_t)d * HEADS + h];
  const float4 a = ((const float4*)(xl + (size_t)s * HID))[lane];
  float* o = out + (size_t)d * HID + lane * 4;
  atomicAdd(o + 0, alpha * a.x);
  atomicAdd(o + 1, alpha * a.y);
  atomicAdd(o + 2, alpha * a.z);
  atomicAdd(o + 3, alpha * a.w);
}

__global__ void l1_elu_kernel(float* __restrict__ h, const float* __restrict__ bias, int total) {
  const int i = blockIdx.x * blockDim.x + threadIdx.x;
  if (i >= total) return;
  const float v = h[i] + bias[i & (HID - 1)];
  h[i] = (v > 0.f) ? v : expm1f(v);
}

// ---------- layer 2 (H=1, C=64): wave per edge ----------
__global__ void l2_logits_kernel(const float* __restrict__ hl, const float* __restrict__ hr,
                                 const float* __restrict__ w2e, const float* __restrict__ att2,
                                 const int* __restrict__ src, const int* __restrict__ dst,
                                 const float* __restrict__ ew, const float* __restrict__ mean_attr,
                                 float* __restrict__ logits, float* __restrict__ lmax,
                                 int E, int Etot) {
  const int lane = threadIdx.x & 31;
  const int edge = blockIdx.x * (blockDim.x >> 5) + (threadIdx.x >> 5);
  if (edge >= Etot) return;
  int s, d; float attr;
  if (edge < E) { s = ntload(src + edge); d = ntload(dst + edge); attr = ntload(ew + edge); }
  else          { s = edge - E;  d = s;   attr = mean_attr[s]; }
  const float2 a  = ((const float2*)(hl + (size_t)s * DIM_OUT))[lane];
  const float2 b  = ((const float2*)(hr + (size_t)d * DIM_OUT))[lane];
  const float2 e2 = ((const float2*)w2e)[lane];
  const float2 at = ((const float2*)att2)[lane];
  float m0 = a.x + b.x + e2.x * attr; m0 = (m0 > 0.f) ? m0 : 0.2f * m0;
  float m1 = a.y + b.y + e2.y * attr; m1 = (m1 > 0.f) ? m1 : 0.2f * m1;
  float p = m0 * at.x + m1 * at.y;
  #pragma unroll
  for (int o = 1; o < 32; o <<= 1) p += __shfl_xor(p, o, 32);
  if (lane == 0) {
    ntstore(logits + edge, p);
    atomicMaxFloat(&lmax[d], p);
  }
}

__global__ void l2_exp_kernel(float* __restrict__ logits, const float* __restrict__ lmax,
                              float* __restrict__ denom, const int* __restrict__ dst,
                              int E, int Etot) {
  const int e = blockIdx.x * blockDim.x + threadIdx.x;
  if (e >= Etot) return;
  const int d = (e < E) ? ntload(dst + e) : (e - E);
  const float ex = expf(ntload(logits + e) - lmax[d]);
  ntstore(logits + e, ex);
  atomicAdd(&denom[d], ex);
}

__global__ void l2_aggregate_kernel(const float* __restrict__ hl, const float* __restrict__ logits,
                                    const float* __restrict__ denom,
                                    const int* __restrict__ src, const int* __restrict__ dst,
                                    float* __restrict__ out, int E, int Etot) {
  const int lane = threadIdx.x & 31;
  const int edge = blockIdx.x * (blockDim.x >> 5) + (threadIdx.x >> 5);
  if (edge >= Etot) return;
  int s, d;
  if (edge < E) { s = ntload(src + edge); d = ntload(dst + edge); } else { s = edge - E; d = s; }
  const float alpha = ntload(logits + edge) / denom[d];
  const float2 a = ((const float2*)(hl + (size_t)s * DIM_OUT))[lane];
  float* o = out + (size_t)d * DIM_OUT + lane * 2;
  atomicAdd(o + 0, alpha * a.x);
  atomicAdd(o + 1, alpha * a.y);
}

// ---------- epilogue: bias + log_softmax + fc ----------
__global__ void final_kernel(const float* __restrict__ acc2, const float* __restrict__ bias2,
                             const float* __restrict__ fcw, const float* __restrict__ fcb,
                             float* __restrict__ out, int N) {
  const int lane = threadIdx.x & 31;
  const int node = blockIdx.x * (blockDim.x >> 5) + (threadIdx.x >> 5);
  if (node >= N) return;
  float2 v = ((const float2*)(acc2 + (size_t)node * DIM_OUT))[lane];
  const float2 b = ((const float2*)bias2)[lane];
  v.x += b.x; v.y += b.y;
  float mx = fmaxf(v.x, v.y);
  #pragma unroll
  for (int o = 1; o < 32; o <<= 1) mx = fmaxf(mx, __shfl_xor(mx, o, 32));
  float se = expf(v.x - mx) + expf(v.y - mx);
  #pragma unroll
  for (int o = 1; o < 32; o <<= 1) se += __shfl_xor(se, o, 32);
  const float ls = mx + logf(se);
  const float2 w = ((const float2*)fcw)[lane];
  float p = (v.x - ls) * w.x + (v.y - ls) * w.y;
  #pragma unroll
  for (int o = 1; o < 32; o <<= 1) p += __shfl_xor(p, o, 32);
  if (lane == 0) out[node] = p + fcb[0];
}

// ---------- launch ----------
extern "C" void kernel_launch(void* const* d_in, const int* in_sizes, int n_in,
                              void* d_out, int out_size, void* d_ws, size_t ws_size,
                              hipStream_t stream) {
  const float* x    = (const float*)d_in[0];
  const int*   eidx = (const int*)  d_in[1];
  const float* ew   = (const float*)d_in[2];
  const float* w1l  = (const float*)d_in[3];
  const float* b1l  = (const float*)d_in[4];
  const float* w1r  = (const float*)d_in[5];
  const float* b1r  = (const float*)d_in[6];
  const float* w1e  = (const float*)d_in[7];
  const float* att1 = (const float*)d_in[8];
  const float* bias1= (const float*)d_in[9];
  const float* w2l  = (const float*)d_in[10];
  const float* b2l  = (const float*)d_in[11];
  const float* w2r  = (const float*)d_in[12];
  const float* b2r  = (const float*)d_in[13];
  const float* w2e  = (const float*)d_in[14];
  const float* att2 = (const float*)d_in[15];
  const float* bias2= (const float*)d_in[16];
  const float* fcw  = (const float*)d_in[17];
  const float* fcb  = (const float*)d_in[18];
  float* out = (float*)d_out;

  const int N = in_sizes[0] / DIM_IN;
  const int E = in_sizes[1] / 2;
  const int Etot = E + N;
  const int* src = eidx;
  const int* dst = eidx + E;

  // workspace layout (floats)
  float* ws = (float*)d_ws;
  float* xl1   = ws; ws += (size_t)N * HID;
  float* xr1   = ws; ws += (size_t)N * HID;
  float* acc1  = ws; ws += (size_t)N * HID;      // becomes h after ELU (in place)
  float* hl2   = ws; ws += (size_t)N * DIM_OUT;
  float* hr2   = ws; ws += (size_t)N * DIM_OUT;
  float* acc2  = ws; ws += (size_t)N * DIM_OUT;
  float* cnt   = ws; ws += N;
  float* sm    = ws; ws += N;                    // cnt+sm contiguous
  float* meanA = ws; ws += N;
  float* lmax1 = ws; ws += (size_t)N * HEADS;
  float* den1  = ws; ws += (size_t)N * HEADS;
  float* lmax2 = ws; ws += N;
  float* den2  = ws; ws += N;
  float* lg1   = ws; ws += (size_t)Etot * HEADS;
  float* lg2   = ws; ws += Etot;

  const int B = 256;
  const float NINF = -INFINITY;
  #define CDIV(a, b) (((a) + (b) - 1) / (b))

  // init accumulators / maxima
  fill_kernel<<<CDIV(N * HID, B), B, 0, stream>>>(acc1, 0.f, N * HID);
  fill_kernel<<<CDIV(N * DIM_OUT, B), B, 0, stream>>>(acc2, 0.f, N * DIM_OUT);
  fill_kernel<<<CDIV(2 * N, B), B, 0, stream>>>(cnt, 0.f, 2 * N);
  fill_kernel<<<CDIV(N * HEADS, B), B, 0, stream>>>(lmax1, NINF, N * HEADS);
  fill_kernel<<<CDIV(N * HEADS, B), B, 0, stream>>>(den1, 0.f, N * HEADS);
  fill_kernel<<<CDIV(N, B), B, 0, stream>>>(lmax2, NINF, N);
  fill_kernel<<<CDIV(N, B), B, 0, stream>>>(den2, 0.f, N);

  // self-loop mean edge attr
  degree_kernel<<<CDIV(E, B), B, 0, stream>>>(dst, ew, cnt, sm, E);
  mean_kernel<<<CDIV(N, B), B, 0, stream>>>(cnt, sm, meanA, N);

  // layer-1 GEMMs (fp32 WMMA)
  gemm_wmma_kernel<DIM_IN, HID><<<CDIV(N, 16), 32 * (HID / 16), 0, stream>>>(x, w1l, b1l, xl1, N);
  gemm_wmma_kernel<DIM_IN, HID><<<CDIV(N, 16), 32 * (HID / 16), 0, stream>>>(x, w1r, b1r, xr1, N);

  // layer-1 edge passes (wave per edge; 8 edges per 256-thread block)
  const int gE = CDIV(Etot, 8);
  l1_logits_kernel<<<gE, B, 0, stream>>>(xl1, xr1, w1e, att1, src, dst, ew, meanA, lg1, lmax1, E, Etot);
  l1_exp_kernel<<<CDIV(Etot * HEADS, B), B, 0, stream>>>(lg1, lmax1, den1, dst, E, Etot);
  l1_aggregate_kernel<<<gE, B, 0, stream>>>(xl1, lg1, den1, src, dst, acc1, E, Etot);
  l1_elu_kernel<<<CDIV(N * HID, B), B, 0, stream>>>(acc1, bias1, N * HID);

  // layer-2 GEMMs (fp32 WMMA)
  gemm_wmma_kernel<HID, DIM_OUT><<<CDIV(N, 16), 32 * (DIM_OUT / 16), 0, stream>>>(acc1, w2l, b2l, hl2, N);
  gemm_wmma_kernel<HID, DIM_OUT><<<CDIV(N, 16), 32 * (DIM_OUT / 16), 0, stream>>>(acc1, w2r, b2r, hr2, N);

  // layer-2 edge passes
  l2_logits_kernel<<<gE, B, 0, stream>>>(hl2, hr2, w2e, att2, src, dst, ew, meanA, lg2, lmax2, E, Etot);
  l2_exp_kernel<<<CDIV(Etot, B), B, 0, stream>>>(lg2, lmax2, den2, dst, E, Etot);
  l2_aggregate_kernel<<<gE, B, 0, stream>>>(hl2, lg2, den2, src, dst, acc2, E, Etot);

  // bias + log_softmax + fc
  final_kernel<<<CDIV(N, 8), B, 0, stream>>>(acc2, bias2, fcw, fcb, out, N);

  (void)n_in; (void)out_size; (void)ws_size;
}